// NodeGraphConv_46170898432062
// MI455X (gfx1250) — compile-verified
//
#include <hip/hip_runtime.h>
#include <hip/hip_bf16.h>

typedef float v2f __attribute__((ext_vector_type(2)));
typedef float v8f __attribute__((ext_vector_type(8)));

#define NNODES 100000
#define NEDGES 1600000
#define FEAT   128

// ---------------------------------------------------------------------------
// Zero int region (degree counters + bucket cursors).
// ---------------------------------------------------------------------------
__global__ __launch_bounds__(256) void zero_i32_kernel(int* __restrict__ p, int n) {
    int i = blockIdx.x * 256 + threadIdx.x;
    if (i < n) p[i] = 0;
}

// ---------------------------------------------------------------------------
// CSR build step 1: in-degree histogram (1.6M int atomics, replaces the 614M
// f32 atomics a scatter formulation would need).
// ---------------------------------------------------------------------------
__global__ __launch_bounds__(256) void count_deg_kernel(
    const int* __restrict__ ei, int* __restrict__ deg, int ne)
{
    int e = blockIdx.x * 256 + threadIdx.x;
    if (e >= ne) return;
    atomicAdd(&deg[ei[ne + e]], 1);     // dst = edge_index[1][e]
}

// ---------------------------------------------------------------------------
// CSR build step 2: exclusive prefix sum over deg[0..n) -> off[0..n].
// Single 1024-thread workgroup, chunked Hillis-Steele scan in LDS with a
// running carry. n=100k -> 98 chunks; negligible vs the atomics it saves.
// ---------------------------------------------------------------------------
__global__ __launch_bounds__(1024) void scan_kernel(
    const int* __restrict__ deg, int* __restrict__ off, int n)
{
    __shared__ int smem[1024];
    __shared__ int carry;
    const int tid = threadIdx.x;
    if (tid == 0) carry = 0;
    __syncthreads();
    for (int base = 0; base < n; base += 1024) {
        const int i = base + tid;
        const int v = (i < n) ? deg[i] : 0;
        smem[tid] = v;
        __syncthreads();
        #pragma unroll
        for (int d = 1; d < 1024; d <<= 1) {
            int t = (tid >= d) ? smem[tid - d] : 0;
            __syncthreads();
            smem[tid] += t;
            __syncthreads();
        }
        const int incl = smem[tid];
        const int c = carry;
        if (i < n) off[i] = c + incl - v;     // exclusive
        __syncthreads();
        if (tid == 1023) carry = c + smem[1023];
        __syncthreads();
    }
    if (tid == 0) off[n] = carry;             // == total edge count
}

// ---------------------------------------------------------------------------
// CSR build step 3: bucket fill. csr[slot] = src for each edge grouped by dst.
// ---------------------------------------------------------------------------
__global__ __launch_bounds__(256) void fill_csr_kernel(
    const int* __restrict__ ei, const int* __restrict__ off,
    int* __restrict__ cursor, int* __restrict__ csr, int ne)
{
    int e = blockIdx.x * 256 + threadIdx.x;
    if (e >= ne) return;
    const int dst  = ei[ne + e];
    const int slot = off[dst] + atomicAdd(&cursor[dst], 1);
    csr[slot] = ei[e];                        // src = edge_index[0][e]
}

// ---------------------------------------------------------------------------
// Aggregation as CSR gather: one wave32 per node, each lane owns one float4
// column chunk. Neighbor indices are fetched 32-at-a-time with ONE coalesced
// load (csr[base+lane]) and broadcast to the wave via __shfl (VALU lane
// permute, no memory) -- removing the serial index-load/row-load dependency
// chain. Row reads are coalesced 512B and L2-resident (h=51MB vs 192MB L2).
// Zero f32 atomics; agg written exactly once.
// ---------------------------------------------------------------------------
__global__ __launch_bounds__(256) void gather_agg_kernel(
    const int* __restrict__ off, const int* __restrict__ csr,
    const float* __restrict__ h, float* __restrict__ agg, int nnodes)
{
    int gtid = blockIdx.x * 256 + threadIdx.x;
    int node = gtid >> 5;
    int lane = threadIdx.x & 31;
    if (node >= nnodes) return;
    const int beg = off[node];
    const int end = off[node + 1];
    float4 acc = make_float4(0.f, 0.f, 0.f, 0.f);
    for (int base = beg; base < end; base += 32) {
        const int cnt  = min(32, end - base);
        const int sidx = (base + lane < end) ? csr[base + lane] : 0;
        for (int i = 0; i < cnt; ++i) {
            const int s = __shfl(sidx, i, 32);        // broadcast lane i
            const float4 v = *(const float4*)(h + (long)s * FEAT + lane * 4);
            acc.x += v.x; acc.y += v.y; acc.z += v.z; acc.w += v.w;
        }
    }
    *(float4*)(agg + (long)node * FEAT + lane * 4) = acc;
}

// ---------------------------------------------------------------------------
// Fused dual GEMM:  out = relu?( A1@W1 + A2@W2 + bias )
//
// One wave owns MT 16-row M tiles x NT 16-col N tiles (MT*NT accumulators).
// 32-bit element offsets off SGPR kernarg bases -> saddr+voffset+imm loads.
// Software-pipelined: fragments for k+4 load into a 'next' buffer while the
// 2*MT*NT WMMAs for k consume the 'current' buffer (last k-step peeled), so
// loads issue a full 16-WMMA step ahead of their loadcnt wait.
//
// f32 16x16x4 fragment layout (ISA 7.12.2):
//   A: v0/v1; lanes 0-15: row=lane, k={k,k+1}; lanes 16-31: row=lane-16, k={k+2,k+3}
//   B: v0/v1; lanes 0-15: col=lane, k={k,k+1}; lanes 16-31: col=lane-16, k={k+2,k+3}
//   C/D: 8 VGPRs; element r -> (row = tile*16 + r + 8*(lane>=16), col as B)
// ---------------------------------------------------------------------------
template<int COUT, int MT, int NT, bool TAIL>
__device__ __forceinline__ void load_frags(
    const float* __restrict__ A1, const float* __restrict__ A2,
    const float* __restrict__ W1, const float* __restrict__ W2,
    unsigned ao, const unsigned* bo, const bool* ok, int k,
    v2f* a1, v2f* a2, v2f* b1, v2f* b2)
{
    #pragma unroll
    for (int mt = 0; mt < MT; ++mt) {
        a1[mt] = *(const v2f*)(A1 + ao + (mt * 16 * FEAT + k));
        a2[mt] = *(const v2f*)(A2 + ao + (mt * 16 * FEAT + k));
    }
    #pragma unroll
    for (int nt = 0; nt < NT; ++nt) {
        b1[nt].x = W1[bo[nt] + (k)     * COUT];
        b1[nt].y = W1[bo[nt] + (k + 1) * COUT];
        b2[nt].x = W2[bo[nt] + (k)     * COUT];
        b2[nt].y = W2[bo[nt] + (k + 1) * COUT];
        if (TAIL && !ok[nt]) { b1[nt].x = b1[nt].y = b2[nt].x = b2[nt].y = 0.0f; }
    }
}

template<int COUT, int MT, int NT, bool RELU, bool TAIL>
__global__ __launch_bounds__(32) void gemm_dual_wmma_kernel(
    const float* __restrict__ A1,   // [N, 128] aggregated messages
    const float* __restrict__ A2,   // [N, 128] node features
    const float* __restrict__ W1,   // [128, COUT] lin_rel
    const float* __restrict__ W2,   // [128, COUT] lin_root
    const float* __restrict__ bias, // [COUT]
    float* __restrict__ out)        // [N, COUT]
{
    const int lane = threadIdx.x;          // wave32, EXEC all ones at WMMAs
    const int half = lane >> 4;
    const int l16  = lane & 15;
    const unsigned rowBase = blockIdx.y * (MT * 16);
    const unsigned colBase = blockIdx.x * (NT * 16);

    const unsigned ao = (rowBase + l16) * FEAT + 2 * half;

    unsigned bo[NT];
    bool     ok[NT];
    #pragma unroll
    for (int nt = 0; nt < NT; ++nt) {
        const unsigned col = colBase + nt * 16 + l16;
        ok[nt] = !TAIL || (col < (unsigned)COUT);   // folds for TAIL=false
        const unsigned cc = ok[nt] ? col : (unsigned)(COUT - 1);
        bo[nt] = (unsigned)(2 * half) * COUT + cc;
    }

    v8f acc[MT][NT];
    #pragma unroll
    for (int mt = 0; mt < MT; ++mt)
        #pragma unroll
        for (int nt = 0; nt < NT; ++nt) acc[mt][nt] = (v8f){};

    v2f a1c[MT], a2c[MT], b1c[NT], b2c[NT];
    v2f a1n[MT], a2n[MT], b1n[NT], b2n[NT];
    load_frags<COUT, MT, NT, TAIL>(A1, A2, W1, W2, ao, bo, ok, 0,
                                   a1c, a2c, b1c, b2c);

    #pragma unroll 4
    for (int k = 0; k < FEAT - 4; k += 4) {
        load_frags<COUT, MT, NT, TAIL>(A1, A2, W1, W2, ao, bo, ok, k + 4,
                                       a1n, a2n, b1n, b2n);
        #pragma unroll
        for (int nt = 0; nt < NT; ++nt)
            #pragma unroll
            for (int mt = 0; mt < MT; ++mt) {
                acc[mt][nt] = __builtin_amdgcn_wmma_f32_16x16x4_f32(
                    false, a1c[mt], false, b1c[nt], (short)0, acc[mt][nt], false, false);
                acc[mt][nt] = __builtin_amdgcn_wmma_f32_16x16x4_f32(
                    false, a2c[mt], false, b2c[nt], (short)0, acc[mt][nt], false, false);
            }
        #pragma unroll
        for (int mt = 0; mt < MT; ++mt) { a1c[mt] = a1n[mt]; a2c[mt] = a2n[mt]; }
        #pragma unroll
        for (int nt = 0; nt < NT; ++nt) { b1c[nt] = b1n[nt]; b2c[nt] = b2n[nt]; }
    }
    // peeled last k-step
    #pragma unroll
    for (int nt = 0; nt < NT; ++nt)
        #pragma unroll
        for (int mt = 0; mt < MT; ++mt) {
            acc[mt][nt] = __builtin_amdgcn_wmma_f32_16x16x4_f32(
                false, a1c[mt], false, b1c[nt], (short)0, acc[mt][nt], false, false);
            acc[mt][nt] = __builtin_amdgcn_wmma_f32_16x16x4_f32(
                false, a2c[mt], false, b2c[nt], (short)0, acc[mt][nt], false, false);
        }

    #pragma unroll
    for (int mt = 0; mt < MT; ++mt) {
        #pragma unroll
        for (int nt = 0; nt < NT; ++nt) {
            if (ok[nt]) {
                const unsigned col = colBase + nt * 16 + l16;
                const float bv = bias[col];
                #pragma unroll
                for (int r = 0; r < 8; ++r) {
                    float v = acc[mt][nt][r] + bv;
                    if (RELU) v = fmaxf(v, 0.0f);
                    out[(rowBase + mt * 16 + r + 8 * half) * COUT + col] = v;
                }
            }
        }
    }
}

// ---------------------------------------------------------------------------
// Launch: CSR build (once) + 3x [gather, dual-WMMA GEMM].
// ws layout: [agg | h1 | h2 | deg | cursor | off | csr]  (~161 MB)
// ---------------------------------------------------------------------------
extern "C" void kernel_launch(void* const* d_in, const int* in_sizes, int n_in,
                              void* d_out, int out_size, void* d_ws, size_t ws_size,
                              hipStream_t stream) {
    const float* x   = (const float*)d_in[0];
    const int*   ei  = (const int*)d_in[1];
    const float* Wr0 = (const float*)d_in[2];
    const float* Ws0 = (const float*)d_in[3];
    const float* b0  = (const float*)d_in[4];
    const float* Wr1 = (const float*)d_in[5];
    const float* Ws1 = (const float*)d_in[6];
    const float* b1  = (const float*)d_in[7];
    const float* Wr2 = (const float*)d_in[8];
    const float* Ws2 = (const float*)d_in[9];
    const float* b2  = (const float*)d_in[10];
    float* out = (float*)d_out;

    float* agg    = (float*)d_ws;
    float* h1     = agg + (long)NNODES * FEAT;
    float* h2     = h1  + (long)NNODES * FEAT;
    int*   deg    = (int*)(h2 + (long)NNODES * FEAT);
    int*   cursor = deg + NNODES;
    int*   off    = cursor + NNODES;          // NNODES+1 entries
    int*   csr    = off + NNODES + 1;         // NEDGES entries

    dim3 igrid((2 * NNODES + 255) / 256);               // zero deg+cursor
    dim3 egrid((NEDGES + 255) / 256);                   // per-edge kernels
    dim3 agrid(((long)NNODES * 32 + 255) / 256);        // 1 wave per node
    dim3 ggrid(FEAT / 64, NNODES / 32);                 // 2 x 3125 (MT=2, NT=4)
    dim3 ogrid(1, NNODES / 32);                         // 1 x 3125 (MT=2, NT=3)

    // ---- Build CSR (edge structure is layer-invariant) ----
    zero_i32_kernel<<<igrid, 256, 0, stream>>>(deg, 2 * NNODES);   // deg+cursor
    count_deg_kernel<<<egrid, 256, 0, stream>>>(ei, deg, NEDGES);
    scan_kernel<<<1, 1024, 0, stream>>>(deg, off, NNODES);
    fill_csr_kernel<<<egrid, 256, 0, stream>>>(ei, off, cursor, csr, NEDGES);

    // Layer 0: h1 = relu(segsum(x) @ Wr0 + b0 + x @ Ws0)
    gather_agg_kernel<<<agrid, 256, 0, stream>>>(off, csr, x, agg, NNODES);
    gemm_dual_wmma_kernel<128, 2, 4, true, false><<<ggrid, 32, 0, stream>>>(
        agg, x, Wr0, Ws0, b0, h1);

    // Layer 1: h2 = relu(segsum(h1) @ Wr1 + b1 + h1 @ Ws1)
    gather_agg_kernel<<<agrid, 256, 0, stream>>>(off, csr, h1, agg, NNODES);
    gemm_dual_wmma_kernel<128, 2, 4, true, false><<<ggrid, 32, 0, stream>>>(
        agg, h1, Wr1, Ws1, b1, h2);

    // Layer 2: out = segsum(h2) @ Wr2 + b2 + h2 @ Ws2   (C=40, no relu)
    gather_agg_kernel<<<agrid, 256, 0, stream>>>(off, csr, h2, agg, NNODES);
    gemm_dual_wmma_kernel<40, 2, 3, false, true><<<ogrid, 32, 0, stream>>>(
        agg, h2, Wr2, Ws2, b2, out);
}